// GIN_19413252178639
// MI455X (gfx1250) — compile-verified
//
#include <hip/hip_runtime.h>
#include <hip/hip_bf16.h>

typedef __attribute__((ext_vector_type(2))) float v2f;
typedef __attribute__((ext_vector_type(8))) float v8f;

#define WPB 4              // waves per block (wave32): 4 waves = 128 threads, 4 node-tiles
static constexpr int HID  = 64;    // hidden channels
static constexpr int HID2 = 128;   // 2*H
static constexpr float BNE = 1e-5f;

// ---------------------------------------------------------------------------
// Kernel 0: zero the workspace prefix (agg + sums + cnt). Graph-capture safe.
// ---------------------------------------------------------------------------
__global__ __launch_bounds__(256) void gin_zero(float* __restrict__ p, long n) {
    long i = (long)blockIdx.x * 256 + threadIdx.x;
    if (i < n) p[i] = 0.0f;
}

// ---------------------------------------------------------------------------
// Kernel 1: edge scatter-add   agg[dst] += x[src]   (the bandwidth-bound phase)
// One thread per edge; x row is 16B so a single b128 load; agg (8MB) stays in L2
// so the 64M f32 atomics resolve at L2 atomic units.
// ---------------------------------------------------------------------------
__global__ __launch_bounds__(256) void gin_edge_agg(const int* __restrict__ src,
                                                    const int* __restrict__ dst,
                                                    const float* __restrict__ x,
                                                    float* __restrict__ agg, int E) {
    int e = blockIdx.x * 256 + threadIdx.x;
    if (e >= E) return;
    int s = src[e];
    int d = dst[e];
    const float4 xs = *(const float4*)(x + (size_t)s * 4);
    float* a = agg + (size_t)d * 4;
    atomicAdd(a + 0, xs.x);
    atomicAdd(a + 1, xs.y);
    atomicAdd(a + 2, xs.z);
    atomicAdd(a + 3, xs.w);
}

// ---------------------------------------------------------------------------
// Kernel 2: fused node MLP (WMMA f32 16x16x4) + BN + ReLU + graph-sum pooling.
// One wave handles 16 nodes.
//  A layout (32-bit 16x4, ISA 7.12.2): lane = row M (lane&15), half-wave picks
//  K pair {0,1} / {2,3} in VGPR0/VGPR1.
//  B layout (mirror): lane = col N, half-wave picks K pair in the two VGPRs.
//  C layout: VGPR r -> row r (+8 for lanes 16..31), lane&15 -> col.
// Layer1 -> LDS (transpose C-layout -> A-layout), Layer2 -> atomic pooling.
// ---------------------------------------------------------------------------
__global__ __launch_bounds__(WPB * 32) void gin_mlp_pool(
    const float* __restrict__ x,  const float* __restrict__ agg,
    const float* __restrict__ epsp,
    const float* __restrict__ w1, const float* __restrict__ b1,
    const float* __restrict__ g1, const float* __restrict__ be1,
    const float* __restrict__ m1, const float* __restrict__ v1,
    const float* __restrict__ w2, const float* __restrict__ b2,
    const float* __restrict__ g2, const float* __restrict__ be2,
    const float* __restrict__ m2, const float* __restrict__ v2,
    const int* __restrict__ batch,
    float* __restrict__ sums, float* __restrict__ cnt,
    int N, int ntiles) {
    // 132-float row stride: 132 % 64 = 4 -> consecutive rows hit rotated banks
    __shared__ __align__(16) float h1s[WPB][16][132];

    const int lane = threadIdx.x & 31;
    const int wave = threadIdx.x >> 5;
    const int tile = blockIdx.x * WPB + wave;
    const bool active = (tile < ntiles);
    const int row  = lane & 15;               // matrix row (A) / col (B,C)
    const int koff = (lane < 16) ? 0 : 2;     // which K pair this half-wave owns
    const int rbase = (lane < 16) ? 0 : 8;    // C row offset for this half-wave

    if (active) {
        const int node0 = tile * 16;
        const float ep1 = 1.0f + epsp[0];
        // ---- build A (16x4): h = (1+eps)*x + agg -----------------------------
        int mload = node0 + row; if (mload >= N) mload = N - 1;   // tail clamp
        v2f av1;
        {
            const float2 xv = *(const float2*)(x   + (size_t)mload * 4 + koff);
            const float2 gv = *(const float2*)(agg + (size_t)mload * 4 + koff);
            av1.x = ep1 * xv.x + gv.x;
            av1.y = ep1 * xv.y + gv.y;
        }
        // ---- layer 1: [16x4] @ [4x128] -> 8 col-tiles, ONE wmma each (K=4) ---
        for (int t = 0; t < 8; ++t) {
            const int n = t * 16 + row;       // output column owned by this lane
            v2f bv;
            bv.x = w1[(size_t)koff       * HID2 + n];
            bv.y = w1[(size_t)(koff + 1) * HID2 + n];
            v8f c = {};
            c = __builtin_amdgcn_wmma_f32_16x16x4_f32(
                    false, av1, false, bv, (short)0, c, false, false);
            // fold bias+BN: (z + b1 - m1)*s + be1 ; then ReLU; write transposed
            const float s  = g1[n] * rsqrtf(v1[n] + BNE);
            const float sh = (b1[n] - m1[n]) * s + be1[n];
#pragma unroll
            for (int r = 0; r < 8; ++r) {
                const float hv = fmaxf(c[r] * s + sh, 0.0f);
                h1s[wave][r + rbase][t * 16 + row] = hv;
            }
        }
    }
    __syncthreads();   // uniform across block (inactive waves participate)
    if (active) {
        const int node0 = tile * 16;
        // per-row graph ids, reused across all col tiles
        int gb[8];
#pragma unroll
        for (int r = 0; r < 8; ++r) {
            int nd = node0 + r + rbase; if (nd >= N) nd = N - 1;
            gb[r] = batch[nd];
        }
        // ---- layer 2: [16x128] @ [128x64] -> 4 col-tiles x 32 K-steps --------
        for (int t2 = 0; t2 < 4; ++t2) {
            const int n = t2 * 16 + row;
            v8f c = {};
#pragma unroll 4
            for (int k = 0; k < HID2; k += 4) {
                const v2f av = *(const v2f*)&h1s[wave][row][k + koff];
                v2f bv;
                bv.x = w2[(size_t)(k + koff)     * HID + n];
                bv.y = w2[(size_t)(k + koff + 1) * HID + n];
                c = __builtin_amdgcn_wmma_f32_16x16x4_f32(
                        false, av, false, bv, (short)0, c, false, false);
            }
            const float s  = g2[n] * rsqrtf(v2[n] + BNE);
            const float sh = (b2[n] - m2[n]) * s + be2[n];
#pragma unroll
            for (int r = 0; r < 8; ++r) {
                const int nd = node0 + r + rbase;
                if (nd < N) {
                    const float hv = fmaxf(c[r] * s + sh, 0.0f);
                    atomicAdd(&sums[(size_t)gb[r] * HID + n], hv);
                }
            }
        }
        if (lane < 16 && (node0 + lane) < N)
            atomicAdd(&cnt[batch[node0 + lane]], 1.0f);
    }
}

// ---------------------------------------------------------------------------
// Kernel 3: per-graph head: pooled = sums/max(cnt,1); logits = pooled@w3 + b3;
// log_softmax over 2 classes. One thread per graph (G=5000).
// ---------------------------------------------------------------------------
__global__ __launch_bounds__(256) void gin_head(const float* __restrict__ sums,
                                                const float* __restrict__ cnt,
                                                const float* __restrict__ w3,
                                                const float* __restrict__ b3,
                                                float* __restrict__ out, int G) {
    int g = blockIdx.x * 256 + threadIdx.x;
    if (g >= G) return;
    const float inv = 1.0f / fmaxf(cnt[g], 1.0f);
    float l0 = b3[0], l1 = b3[1];
    const float* srow = sums + (size_t)g * HID;
#pragma unroll 8
    for (int k = 0; k < HID; ++k) {
        const float p = srow[k] * inv;
        l0 = fmaf(p, w3[k * 2 + 0], l0);
        l1 = fmaf(p, w3[k * 2 + 1], l1);
    }
    const float mx  = fmaxf(l0, l1);
    const float lse = mx + logf(expf(l0 - mx) + expf(l1 - mx));
    out[g * 2 + 0] = l0 - lse;
    out[g * 2 + 1] = l1 - lse;
}

// ---------------------------------------------------------------------------
extern "C" void kernel_launch(void* const* d_in, const int* in_sizes, int n_in,
                              void* d_out, int out_size, void* d_ws, size_t ws_size,
                              hipStream_t stream) {
    const float* x   = (const float*)d_in[0];
    const float* eps = (const float*)d_in[1];
    const float* w1  = (const float*)d_in[2];
    const float* b1  = (const float*)d_in[3];
    const float* g1  = (const float*)d_in[4];
    const float* be1 = (const float*)d_in[5];
    const float* m1  = (const float*)d_in[6];
    const float* v1  = (const float*)d_in[7];
    const float* w2  = (const float*)d_in[8];
    const float* b2  = (const float*)d_in[9];
    const float* g2  = (const float*)d_in[10];
    const float* be2 = (const float*)d_in[11];
    const float* m2  = (const float*)d_in[12];
    const float* v2  = (const float*)d_in[13];
    const float* w3  = (const float*)d_in[14];
    const float* b3  = (const float*)d_in[15];
    const int* eidx  = (const int*)d_in[16];
    const int* batch = (const int*)d_in[17];

    const int N = in_sizes[0] / 4;
    const int E = in_sizes[16] / 2;
    const int G = out_size / 2;

    // workspace layout: agg[N*4] | sums[G*HID] | cnt[G]
    float* agg  = (float*)d_ws;
    float* sums = agg + (size_t)N * 4;
    float* cnt  = sums + (size_t)G * HID;
    const long zn = (long)N * 4 + (long)G * HID + (long)G;

    gin_zero<<<(int)((zn + 255) / 256), 256, 0, stream>>>((float*)d_ws, zn);

    gin_edge_agg<<<(E + 255) / 256, 256, 0, stream>>>(eidx, eidx + E, x, agg, E);

    const int ntiles = (N + 15) / 16;
    gin_mlp_pool<<<(ntiles + WPB - 1) / WPB, WPB * 32, 0, stream>>>(
        x, agg, eps, w1, b1, g1, be1, m1, v1, w2, b2, g2, be2, m2, v2,
        batch, sums, cnt, N, ntiles);

    gin_head<<<(G + 255) / 256, 256, 0, stream>>>(sums, cnt, w3, b3,
                                                  (float*)d_out, G);
}